// SSM_46231027974765
// MI455X (gfx1250) — compile-verified
//
#include <hip/hip_runtime.h>

typedef __attribute__((ext_vector_type(2))) float v2f;
typedef __attribute__((ext_vector_type(8))) float v8f;

namespace {
constexpr int B_ = 8;      // batch
constexpr int L_ = 8192;   // sequence length
constexpr int H_ = 128;    // input channels
constexpr int P_ = 256;    // state size
constexpr int O_ = 128;    // output channels
constexpr int LC = 64;     // chunk length (time steps per workgroup)
constexpr int NC = L_ / LC;   // 128 chunks
constexpr int TT = LC / 16;   // 4 t-tiles per chunk
}

// D = A(16x4 f32) * B(4x16 f32) + C(16x16 f32), full fp32 precision.
static __device__ __forceinline__ v8f wmma_f32(v2f a, v2f b, v8f c) {
  return __builtin_amdgcn_wmma_f32_16x16x4_f32(
      /*neg_a=*/false, a, /*neg_b=*/false, b,
      /*c_mod=*/(short)0, c, /*reuse_a=*/false, /*reuse_b=*/false);
}

// 32-bit LDS byte address (as3 ptrtoint) for async-to-LDS destinations.
static __device__ __forceinline__ unsigned lds_addr_of(const void* p) {
  return (unsigned)(unsigned long long)(__attribute__((address_space(3))) const void*)p;
}

// ---------------------------------------------------------------------------
// Kernel 1: discretization + operand re-layout (tiny, 1 block of 256 threads).
//   lam      = -|L0| + i|L1| ;  Lbar = exp(lam * exp(log_step))
//   coef     = (Lbar - 1)/lam
//   B-operands stored K-PAIRED for single-b64 WMMA fragment loads:
//     BbP[k>>1][p][k&1]  = (coef[p] * Bc[p][k])        (k over H_)
//     CTp[p>>1][o][p&1]  = C_re[o][p] ; CTn paired = -C_im[o][p]
//   pow[k][p]  = Lbar^(k+1)  k=0..LC-1 ;  Ap[p] = Lbar^LC
// ---------------------------------------------------------------------------
__global__ void SSM_setup_kernel(const float* __restrict__ Lambda,
                                 const float* __restrict__ log_step,
                                 const float* __restrict__ Bm,
                                 const float* __restrict__ Cm,
                                 float* BbP_re, float* BbP_im,
                                 float* CTp_re, float* CTpn_im,
                                 float* Lb_re,  float* Lb_im,
                                 float* Ap_re,  float* Ap_im,
                                 float* pow_re, float* pow_im) {
  int p = threadIdx.x;
  if (p >= P_) return;
  float lr = -fabsf(Lambda[2 * p + 0]);
  float li =  fabsf(Lambda[2 * p + 1]);
  float dt = expf(log_step[p]);
  float mag = expf(lr * dt);
  float ang = li * dt;
  float Lr = mag * cosf(ang);
  float Li = mag * sinf(ang);
  Lb_re[p] = Lr; Lb_im[p] = Li;
  // coef = (Lbar - 1)/lam  (complex division)
  float nr = Lr - 1.0f, ni = Li;
  float den = lr * lr + li * li;
  float cr = (nr * lr + ni * li) / den;
  float ci = (ni * lr - nr * li) / den;
  for (int h = 0; h < H_; ++h) {
    float br = Bm[(p * H_ + h) * 2 + 0];
    float bi = Bm[(p * H_ + h) * 2 + 1];
    const size_t idx = ((size_t)(h >> 1) * P_ + p) * 2 + (h & 1);
    BbP_re[idx] = cr * br - ci * bi;
    BbP_im[idx] = cr * bi + ci * br;
  }
  for (int o = 0; o < O_; ++o) {
    const size_t idx = ((size_t)(p >> 1) * O_ + o) * 2 + (p & 1);
    CTp_re [idx] =  Cm[(o * P_ + p) * 2 + 0];
    CTpn_im[idx] = -Cm[(o * P_ + p) * 2 + 1];
  }
  float pr = Lr, pi = Li;               // Lbar^1
  for (int k = 0; k < LC; ++k) {
    pow_re[k * P_ + p] = pr;
    pow_im[k * P_ + p] = pi;
    if (k == LC - 1) { Ap_re[p] = pr; Ap_im[p] = pi; }
    float tr = pr * Lr - pi * Li;
    float ti = pr * Li + pi * Lr;
    pr = tr; pi = ti;
  }
}

// ---------------------------------------------------------------------------
// Kernel 2: per-(batch,chunk) input GEMM via WMMA + in-register local scan.
// Signal tile staged with GLOBAL_LOAD_ASYNC_TO_LDS_B128 (ASYNCcnt-tracked,
// no VGPR round-trip). 16 waves; wave w owns the state strip p0 = 16*w.
// Writes zero-init local scan result xl[b][l][p] and chunk-final state.
// ---------------------------------------------------------------------------
__global__ __launch_bounds__(512) void SSM_bu_scan_kernel(
    const float* __restrict__ sig,
    const float* __restrict__ BbP_re, const float* __restrict__ BbP_im,
    const float* __restrict__ Lb_re,  const float* __restrict__ Lb_im,
    float* __restrict__ xl_re, float* __restrict__ xl_im,
    float* __restrict__ xend_re, float* __restrict__ xend_im) {
  __shared__ float U[LC][H_ + 2];                 // padded: bank-conflict-free A reads
  const int c = blockIdx.x, b = blockIdx.y;
  const int tid = threadIdx.x;
  const float* src = sig + ((size_t)b * L_ + (size_t)c * LC) * H_;

  // Async bulk stage: LC*H_ floats = 2048 x 16B granules; 512 threads x 4 rounds.
  // Rows are 128 floats (32 granules, 16B-aligned); +2-float row padding kept.
  {
    const unsigned ubase = lds_addr_of(&U[0][0]);
    #pragma unroll
    for (int r = 0; r < (LC * H_) / (4 * 512); ++r) {
      const int g = tid + r * 512;          // 16B-granule index
      const int t = g >> 5;                 // 32 granules per time row
      const int hc = g & 31;
      const float* gp = src + t * H_ + hc * 4;
      const unsigned la = ubase + (unsigned)(t * (H_ + 2) + hc * 4) * 4u;
      asm volatile("global_load_async_to_lds_b128 %0, %1, off"
                   :: "v"(la), "v"(gp) : "memory");
    }
    asm volatile("s_wait_asynccnt 0" ::: "memory");
  }
  __syncthreads();

  const int wave = tid >> 5, lane = tid & 31;
  const int lhalf = lane >> 4, l15 = lane & 15;
  const int p0 = wave * 16;
  const v2f* BbV_re = (const v2f*)BbP_re;
  const v2f* BbV_im = (const v2f*)BbP_im;

  v8f acc_re[TT], acc_im[TT];
  #pragma unroll
  for (int tt = 0; tt < TT; ++tt) { acc_re[tt] = (v8f)0.0f; acc_im[tt] = (v8f)0.0f; }

  // GEMM1: Bu = U @ BbT   (re & im planes), K = H_ in steps of 4.
  // B fragment = single b64 load thanks to K-paired layout.
  for (int k0 = 0; k0 < H_; k0 += 4) {
    const int kp = ((k0 >> 1) + lhalf) * P_ + p0 + l15;
    v2f bre = BbV_re[kp];
    v2f bim = BbV_im[kp];
    const int ka = k0 + 2 * lhalf;
    #pragma unroll
    for (int tt = 0; tt < TT; ++tt) {
      const int t = tt * 16 + l15;
      v2f a; a[0] = U[t][ka]; a[1] = U[t][ka + 1];
      acc_re[tt] = wmma_f32(a, bre, acc_re[tt]);
      acc_im[tt] = wmma_f32(a, bim, acc_im[tt]);
    }
  }

  // local scan over the chunk (zero initial state), in accumulator registers.
  const float Are = Lb_re[p0 + l15];
  const float Aim = Lb_im[p0 + l15];
  float s_re = 0.0f, s_im = 0.0f;
  #pragma unroll
  for (int tt = 0; tt < TT; ++tt) {
    // pass 1: both halves scan with their own init (half0: carried, half1: 0)
    #pragma unroll
    for (int r = 0; r < 8; ++r) {
      float nre = Are * s_re - Aim * s_im + acc_re[tt][r];
      float nim = Are * s_im + Aim * s_re + acc_im[tt][r];
      acc_re[tt][r] = nre; acc_im[tt][r] = nim;
      s_re = nre; s_im = nim;
    }
    // hand half0's final state (t = tt*16+7) to half1 lanes of the same p.
    float sb_re = __shfl_xor(s_re, 16, 32);
    float sb_im = __shfl_xor(s_im, 16, 32);
    float c_re = lhalf ? sb_re : 0.0f;
    float c_im = lhalf ? sb_im : 0.0f;
    // pass 2: half1 fixup  x[8+r] += Lbar^(r+1) * S0
    #pragma unroll
    for (int r = 0; r < 8; ++r) {
      float tr = Are * c_re - Aim * c_im;
      float ti = Are * c_im + Aim * c_re;
      c_re = tr; c_im = ti;
      acc_re[tt][r] += c_re; acc_im[tt][r] += c_im;
    }
    // tile-final true state lives in half1's element 7; hand it back to half0.
    float f_re = acc_re[tt][7], f_im = acc_im[tt][7];
    float g_re = __shfl_xor(f_re, 16, 32);
    float g_im = __shfl_xor(f_im, 16, 32);
    s_re = lhalf ? 0.0f : g_re;
    s_im = lhalf ? 0.0f : g_im;
  }

  // store local scan result; lanes walk consecutive p -> coalesced.
  const size_t base = ((size_t)b * L_ + (size_t)c * LC) * P_;
  #pragma unroll
  for (int tt = 0; tt < TT; ++tt) {
    #pragma unroll
    for (int r = 0; r < 8; ++r) {
      const int t = tt * 16 + 8 * lhalf + r;
      xl_re[base + (size_t)t * P_ + p0 + l15] = acc_re[tt][r];
      xl_im[base + (size_t)t * P_ + p0 + l15] = acc_im[tt][r];
    }
  }
  if (lhalf) {  // chunk-final local state (half1 lanes hold it)
    const size_t idx = ((size_t)b * NC + c) * P_ + p0 + l15;
    xend_re[idx] = acc_re[TT - 1][7];
    xend_im[idx] = acc_im[TT - 1][7];
  }
}

// ---------------------------------------------------------------------------
// Kernel 3: exclusive scan of chunk carries across NC chunks (per (b,p)).
//   carry[b][c] = state entering chunk c ; state' = Lbar^LC * state + xend.
// ---------------------------------------------------------------------------
__global__ void SSM_carry_kernel(const float* __restrict__ xend_re,
                                 const float* __restrict__ xend_im,
                                 const float* __restrict__ Ap_re,
                                 const float* __restrict__ Ap_im,
                                 float* __restrict__ carry_re,
                                 float* __restrict__ carry_im) {
  int idx = blockIdx.x * blockDim.x + threadIdx.x;
  if (idx >= B_ * P_) return;
  const int p = idx & (P_ - 1);
  const int b = idx >> 8;
  const float ar = Ap_re[p], ai = Ap_im[p];
  float sr = 0.0f, si = 0.0f;
  for (int c = 0; c < NC; ++c) {
    const size_t o = ((size_t)b * NC + c) * P_ + p;
    carry_re[o] = sr; carry_im[o] = si;
    float nr = ar * sr - ai * si + xend_re[o];
    float ni = ar * si + ai * sr + xend_im[o];
    sr = nr; si = ni;
  }
}

// ---------------------------------------------------------------------------
// Kernel 4: carry fixup + output GEMM via WMMA.
//   x[t] = xl[t] + Lbar^(t+1)*carry ;  y = x_re@CT_re + x_im@(-CT_im)
// 8 waves; wave w owns output strip o0 = 16*w; loop over 4 t-tiles.
// ---------------------------------------------------------------------------
__global__ __launch_bounds__(256) void SSM_out_kernel(
    const float* __restrict__ xl_re, const float* __restrict__ xl_im,
    const float* __restrict__ carry_re, const float* __restrict__ carry_im,
    const float* __restrict__ pow_re, const float* __restrict__ pow_im,
    const float* __restrict__ CTp_re, const float* __restrict__ CTpn_im,
    float* __restrict__ y) {
  __shared__ float Xr[16][P_ + 2];
  __shared__ float Xi[16][P_ + 2];
  const int c = blockIdx.x, b = blockIdx.y;
  const int tid = threadIdx.x;
  const int wave = tid >> 5, lane = tid & 31;
  const int lhalf = lane >> 4, l15 = lane & 15;
  const int o0 = wave * 16;
  const size_t xbase = ((size_t)b * L_ + (size_t)c * LC) * P_;
  const size_t cbase = ((size_t)b * NC + c) * P_;
  const size_t ybase = ((size_t)b * L_ + (size_t)c * LC) * O_;
  const v2f* CTv_re = (const v2f*)CTp_re;
  const v2f* CTv_im = (const v2f*)CTpn_im;

  for (int tt = 0; tt < TT; ++tt) {
    __syncthreads();
    // stage 16 corrected time rows into padded LDS
    for (int i = tid; i < 16 * P_; i += 256) {
      const int t = i >> 8;
      const int p = i & (P_ - 1);
      const int tg = tt * 16 + t;
      const float cr = carry_re[cbase + p], ci = carry_im[cbase + p];
      const float pr = pow_re[tg * P_ + p], pi = pow_im[tg * P_ + p];
      Xr[t][p] = xl_re[xbase + (size_t)tg * P_ + p] + (pr * cr - pi * ci);
      Xi[t][p] = xl_im[xbase + (size_t)tg * P_ + p] + (pr * ci + pi * cr);
    }
    __syncthreads();

    v8f acc = (v8f)0.0f;
    for (int k0 = 0; k0 < P_; k0 += 4) {
      const int ka = k0 + 2 * lhalf;
      const int kp = ((k0 >> 1) + lhalf) * O_ + o0 + l15;
      v2f ar, ai;
      ar[0] = Xr[l15][ka];  ar[1] = Xr[l15][ka + 1];
      ai[0] = Xi[l15][ka];  ai[1] = Xi[l15][ka + 1];
      v2f br = CTv_re[kp];
      v2f bi = CTv_im[kp];
      acc = wmma_f32(ar, br, acc);   // + x_re * C_re
      acc = wmma_f32(ai, bi, acc);   // - x_im * C_im
    }
    #pragma unroll
    for (int r = 0; r < 8; ++r) {
      const int t = tt * 16 + 8 * lhalf + r;
      y[ybase + (size_t)t * O_ + o0 + l15] = acc[r];
    }
  }
}

// ---------------------------------------------------------------------------
extern "C" void kernel_launch(void* const* d_in, const int* in_sizes, int n_in,
                              void* d_out, int out_size, void* d_ws, size_t ws_size,
                              hipStream_t stream) {
  const float* sig      = (const float*)d_in[0];  // (B,L,H)
  const float* Lambda   = (const float*)d_in[1];  // (P,2)
  const float* log_step = (const float*)d_in[2];  // (P,)
  const float* Bm       = (const float*)d_in[3];  // (P,H,2)
  const float* Cm       = (const float*)d_in[4];  // (O,P,2)
  float* y = (float*)d_out;                       // (B,L,O)

  float* ws = (float*)d_ws;
  size_t off = 0;
  auto alloc = [&](size_t n) { float* p = ws + off; off += n; return p; };
  float* BbP_re  = alloc((size_t)H_ * P_);
  float* BbP_im  = alloc((size_t)H_ * P_);
  float* CTp_re  = alloc((size_t)P_ * O_);
  float* CTpn_im = alloc((size_t)P_ * O_);
  float* Lb_re   = alloc(P_);
  float* Lb_im   = alloc(P_);
  float* Ap_re   = alloc(P_);
  float* Ap_im   = alloc(P_);
  float* pow_re  = alloc((size_t)LC * P_);
  float* pow_im  = alloc((size_t)LC * P_);
  float* xend_re = alloc((size_t)B_ * NC * P_);
  float* xend_im = alloc((size_t)B_ * NC * P_);
  float* carry_re = alloc((size_t)B_ * NC * P_);
  float* carry_im = alloc((size_t)B_ * NC * P_);
  float* xl_re   = alloc((size_t)B_ * L_ * P_);
  float* xl_im   = alloc((size_t)B_ * L_ * P_);
  (void)ws_size; (void)n_in; (void)in_sizes; (void)out_size;

  SSM_setup_kernel<<<1, 256, 0, stream>>>(Lambda, log_step, Bm, Cm,
      BbP_re, BbP_im, CTp_re, CTpn_im, Lb_re, Lb_im, Ap_re, Ap_im, pow_re, pow_im);

  SSM_bu_scan_kernel<<<dim3(NC, B_), 512, 0, stream>>>(
      sig, BbP_re, BbP_im, Lb_re, Lb_im, xl_re, xl_im, xend_re, xend_im);

  SSM_carry_kernel<<<dim3((B_ * P_ + 255) / 256), 256, 0, stream>>>(
      xend_re, xend_im, Ap_re, Ap_im, carry_re, carry_im);

  SSM_out_kernel<<<dim3(NC, B_), 256, 0, stream>>>(
      xl_re, xl_im, carry_re, carry_im, pow_re, pow_im, CTp_re, CTpn_im, y);
}